// RecurrentGCN_78889959293582
// MI455X (gfx1250) — compile-verified
//
#include <hip/hip_runtime.h>
#include <math.h>

typedef float v2f __attribute__((ext_vector_type(2)));
typedef float v8f __attribute__((ext_vector_type(8)));

#define FDIM 64

// ---------------- zero fill (h accumulator) ----------------
__global__ void k_zero_f4(float4* __restrict__ p, int n4) {
    int i = blockIdx.x * blockDim.x + threadIdx.x;
    if (i < n4) p[i] = make_float4(0.f, 0.f, 0.f, 0.f);
}

// ---------------- GRU weight evolution (EvolveGCN-O) ----------------
// One thread per output element of W' [64x64]; 6 length-64 dot products.
__global__ void k_gru_evolve(const float* __restrict__ W,
                             const float* __restrict__ wih,
                             const float* __restrict__ whh,
                             const float* __restrict__ bih,
                             const float* __restrict__ bhh,
                             float* __restrict__ Wev) {
    int t = blockIdx.x * blockDim.x + threadIdx.x;   // 0..4095
    int i = t >> 6, j = t & 63;
    const float* wr  = W   + i * FDIM;
    const float* ir  = wih + j * FDIM;
    const float* iz  = wih + (64 + j) * FDIM;
    const float* inn = wih + (128 + j) * FDIM;
    const float* hr  = whh + j * FDIM;
    const float* hz  = whh + (64 + j) * FDIM;
    const float* hn  = whh + (128 + j) * FDIM;
    float gxr = 0.f, gxz = 0.f, gxn = 0.f, ghr = 0.f, ghz = 0.f, ghn = 0.f;
    for (int k = 0; k < FDIM; ++k) {
        float w = wr[k];
        gxr += w * ir[k];  gxz += w * iz[k];  gxn += w * inn[k];
        ghr += w * hr[k];  ghz += w * hz[k];  ghn += w * hn[k];
    }
    gxr += bih[j];        gxz += bih[64 + j]; gxn += bih[128 + j];
    ghr += bhh[j];        ghz += bhh[64 + j]; ghn += bhh[128 + j];
    float r = 1.f / (1.f + expf(-(gxr + ghr)));
    float z = 1.f / (1.f + expf(-(gxz + ghz)));
    float n = tanhf(gxn + r * ghn);
    Wev[t] = (1.f - z) * n + z * wr[j];
}

// ---------------- xw = x @ W_ev via V_WMMA_F32_16X16X4_F32 ----------------
// One wave per 16x16 output tile; 16 K-steps of 4. f32 in, f32 out (exact).
__global__ void k_gemm_xw(const float* __restrict__ x,
                          const float* __restrict__ Wev,
                          float* __restrict__ xw, int N) {
    const int lane = threadIdx.x & 31;
    const int wv   = threadIdx.x >> 5;
    const int l16  = lane & 15;
    const int half = lane >> 4;

    int tilesM = (N + 15) >> 4;
    int gw = blockIdx.x * (blockDim.x >> 5) + wv;   // global wave id
    if (gw >= tilesM * 4) return;                   // wave-uniform guard

    int tm = gw >> 2;         // M tile index
    int tn = gw & 3;          // N tile index (F=64 -> 4 col tiles)
    int m0 = tm << 4;
    int j0 = tn << 4;

    int arow = m0 + l16; if (arow > N - 1) arow = N - 1;   // clamp, keep EXEC full
    const float* ap = x + (size_t)arow * FDIM;

    v8f c = {0.f, 0.f, 0.f, 0.f, 0.f, 0.f, 0.f, 0.f};
    #pragma unroll
    for (int t = 0; t < 16; ++t) {
        int k = 4 * t + 2 * half;
        v2f a;                                  // A 16x4: lane l16 = row, v0/v1 = k, k+1
        a.x = ap[k];
        a.y = ap[k + 1];
        v2f b;                                  // B 4x16: lanes = col, v0/v1 = rows k, k+1
        b.x = Wev[(size_t)k * FDIM + j0 + l16];
        b.y = Wev[(size_t)(k + 1) * FDIM + j0 + l16];
        c = __builtin_amdgcn_wmma_f32_16x16x4_f32(false, a, false, b,
                                                  (short)0, c, false, false);
    }
    #pragma unroll
    for (int r = 0; r < 8; ++r) {
        int row = m0 + r + 8 * half;            // C layout: vgpr r, halves = M / M+8
        if (row < N) xw[(size_t)row * FDIM + j0 + l16] = c[r];
    }
}

// ---------------- per-edge normalization coefficient ----------------
__global__ void k_norm(const long long* __restrict__ ei,
                       const float* __restrict__ ew,
                       const float* __restrict__ deg,
                       float* __restrict__ nrm, int E) {
    int e = blockIdx.x * blockDim.x + threadIdx.x;
    if (e >= E) return;
    int s = (int)ei[e];
    int d = (int)ei[(size_t)E + e];
    nrm[e] = ew[e] * rsqrtf(deg[s]) * rsqrtf(deg[d]);
}

// ---------------- gather + scatter-add (16 threads/edge, float4 each) -------
__global__ void k_scatter(const long long* __restrict__ ei,
                          const float* __restrict__ nrm,
                          const float* __restrict__ xw,
                          float* __restrict__ h, int E) {
    long long tid = (long long)blockIdx.x * blockDim.x + threadIdx.x;
    int e = (int)(tid >> 4);
    if (e >= E) return;
    int f = ((int)tid & 15) << 2;
    int s = (int)ei[e];
    int d = (int)ei[(size_t)E + e];
    float nv = nrm[e];
    float4 xv = *(const float4*)(xw + (size_t)s * FDIM + f);
    float* hp = h + (size_t)d * FDIM + f;
    unsafeAtomicAdd(hp + 0, nv * xv.x);
    unsafeAtomicAdd(hp + 1, nv * xv.y);
    unsafeAtomicAdd(hp + 2, nv * xv.z);
    unsafeAtomicAdd(hp + 3, nv * xv.w);
}

// ---------------- ReLU + linear readout: one wave32 per node ----------------
__global__ void k_readout(const float* __restrict__ h,
                          const float* __restrict__ lw,
                          const float* __restrict__ lb,
                          float* __restrict__ out, int M) {
    int lane = threadIdx.x & 31;
    int wv   = threadIdx.x >> 5;
    int n = blockIdx.x * (blockDim.x >> 5) + wv;
    if (n >= M) return;
    const float* hp = h + (size_t)n * FDIM;
    float a = hp[lane];       a = a > 0.f ? a : 0.f;
    float b = hp[lane + 32];  b = b > 0.f ? b : 0.f;
    float s = a * lw[lane] + b * lw[lane + 32];
    #pragma unroll
    for (int off = 16; off > 0; off >>= 1) s += __shfl_xor(s, off, 32);
    if (lane == 0) out[n] = s + lb[0];
}

extern "C" void kernel_launch(void* const* d_in, const int* in_sizes, int n_in,
                              void* d_out, int out_size, void* d_ws, size_t ws_size,
                              hipStream_t stream) {
    const float*     x    = (const float*)d_in[0];
    const long long* ei   = (const long long*)d_in[1];   // int64 edge_index [2,E]
    const float*     ew   = (const float*)d_in[2];
    const float*     deg  = (const float*)d_in[3];
    // d_in[4] = target_num scalar; equals out_size, not read
    const float*     W0   = (const float*)d_in[5];
    const float*     wih  = (const float*)d_in[6];
    const float*     whh  = (const float*)d_in[7];
    const float*     bih  = (const float*)d_in[8];
    const float*     bhh  = (const float*)d_in[9];
    const float*     lw   = (const float*)d_in[10];
    const float*     lb   = (const float*)d_in[11];
    float* out = (float*)d_out;

    const int N = in_sizes[3];   // deg has N elements
    const int E = in_sizes[2];   // edge_weight has E elements
    const int M = out_size;      // target_num

    char* ws = (char*)d_ws;
    float* Wev = (float*)ws;                                          // 16 KB
    float* xw  = (float*)(ws + 16384);                                // N*64*4
    float* h   = (float*)(ws + 16384 + (size_t)N * FDIM * 4);         // N*64*4
    float* nrm = (float*)(ws + 16384 + 2 * (size_t)N * FDIM * 4);     // E*4

    // 1) zero accumulator h
    int n4 = (N * FDIM) / 4;
    k_zero_f4<<<(n4 + 255) / 256, 256, 0, stream>>>((float4*)h, n4);

    // 2) evolve W with GRU (tiny)
    k_gru_evolve<<<16, 256, 0, stream>>>(W0, wih, whh, bih, bhh, Wev);

    // 3) xw = x @ W_ev  (WMMA f32, one wave per 16x16 tile)
    int tilesM = (N + 15) / 16;
    int waves  = tilesM * 4;
    k_gemm_xw<<<(waves + 7) / 8, 256, 0, stream>>>(x, Wev, xw, N);

    // 4) per-edge norm coefficients
    k_norm<<<(E + 255) / 256, 256, 0, stream>>>(ei, ew, deg, nrm, E);

    // 5) gather xw[src], scatter-add into h[dst]
    long long sthreads = (long long)E * 16;
    k_scatter<<<(unsigned)((sthreads + 255) / 256), 256, 0, stream>>>(ei, nrm, xw, h, E);

    // 6) ReLU + linear projection
    k_readout<<<(M + 7) / 8, 256, 0, stream>>>(h, lw, lb, out, M);
}